// LM_G2SP_44856638439637
// MI455X (gfx1250) — compile-verified
//
#include <hip/hip_runtime.h>

typedef float v2f __attribute__((ext_vector_type(2)));
typedef float v8f __attribute__((ext_vector_type(8)));

#define A_SIDE 128
#define CCH 64
#define HG 64
#define WG 256
#define BATCH 8
#define PBLK 256
#define NBLK ((A_SIDE * A_SIDE) / PBLK)   // 64 pixel-blocks per batch

__device__ __forceinline__ float wave_sum32(float x) {
#pragma unroll
    for (int m = 16; m >= 1; m >>= 1) x += __shfl_xor(x, m, 32);
    return x;
}

__global__ __launch_bounds__(32)
void lm_init(const float* __restrict__ su, const float* __restrict__ sv,
             const float* __restrict__ hd, float* __restrict__ params) {
    int t = threadIdx.x;
    if (t < BATCH) {
        params[t * 3 + 0] = su[t];
        params[t * 3 + 1] = sv[t];
        params[t * 3 + 2] = hd[t];
    }
}

__global__ __launch_bounds__(PBLK)
void lm_accum(const float* __restrict__ sat, const float* __restrict__ grd,
              const float* __restrict__ conf, const float* __restrict__ camk,
              const float* __restrict__ params, float* __restrict__ partial) {
    const int b   = blockIdx.y;
    const int blk = blockIdx.x;
    const int tid = threadIdx.x;
    const int p   = blk * PBLK + tid;
    const int i   = p >> 7;          // row (v-direction of sat grid)
    const int j   = p & 127;         // col

    // ---- per-batch uniform setup (scalar-ish) ----
    const float su_m  = 20.0f * params[b * 3 + 0];
    const float sv_m  = 20.0f * params[b * 3 + 1];
    const float s10   = 0.17453292519943295f;            // 10/180*pi
    const float theta = params[b * 3 + 2] * s10;
    const float ct = cosf(-theta);
    const float st = sinf(-theta);
    const float* K = camk + b * 9;
    // K rescaled: row0 *= Wg/1024 = 0.25, row1 *= Hg/256 = 0.25. Columns:
    const float k0x = K[0] * 0.25f, k0y = K[3] * 0.25f, k0z = K[6];  // col 0
    const float k1x = K[1] * 0.25f, k1y = K[4] * 0.25f, k1z = K[7];  // col 1
    const float k2x = K[2] * 0.25f, k2y = K[5] * 0.25f, k2z = K[8];  // col 2
    // P columns used (col1 hits Y=0): P0 = ct*c0 + st*c2 ; P2 = -st*c0 + ct*c2
    const float P0x = ct * k0x + st * k2x, P0y = ct * k0y + st * k2y, P0z = ct * k0z + st * k2z;
    const float P2x = -st * k0x + ct * k2x, P2y = -st * k0y + ct * k2y, P2z = -st * k0z + ct * k2z;
    const float P3x = sv_m * k0x + 1.65f * k1x - su_m * k2x;
    const float P3y = sv_m * k0y + 1.65f * k1y - su_m * k2y;
    const float P3z = sv_m * k0z + 1.65f * k1z - su_m * k2z;
    // constant jac direction vectors
    const float kxx = -20.f * k2x, kxy = -20.f * k2y, kxz = -20.f * k2z;   // d/dshift_u
    const float kyx =  20.f * k0x, kyy =  20.f * k0y, kyz =  20.f * k0z;   // d/dshift_v
    const float g0x = s10 * (st * k0x - ct * k2x), g0y = s10 * (st * k0y - ct * k2y), g0z = s10 * (st * k0z - ct * k2z);
    const float g2x = s10 * (ct * k0x + st * k2x), g2y = s10 * (ct * k0y + st * k2y), g2z = s10 * (ct * k0z + st * k2z);

    const float X = 0.8f * (float)(i - 64);
    const float Z = 0.8f * (float)(j - 64);
    const float pu = X * P0x + Z * P2x + P3x;
    const float pv = X * P0y + Z * P2y + P3y;
    const float pw = X * P0z + Z * P2z + P3z;

    float acc[9] = {0.f, 0.f, 0.f, 0.f, 0.f, 0.f, 0.f, 0.f, 0.f};

    bool ok = pw > 1e-6f;
    float iw = 0.f, u = 0.f, v = 0.f;
    if (ok) {
        iw = 1.0f / pw;
        u = pu * iw;
        v = pv * iw;
        ok = (u >= 0.f) && (u <= (float)(WG - 1)) && (v >= 0.f) && (v <= (float)(HG - 1));
    }
    if (ok) {
        // duv rows (per pixel 3x2): duv_n = (t.xy - uv * t.z) / w
        const float t2x = X * g0x + Z * g2x;
        const float t2y = X * g0y + Z * g2y;
        const float t2z = X * g0z + Z * g2z;
        const float du0 = (kxx - u * kxz) * iw, dv0 = (kxy - v * kxz) * iw;
        const float du1 = (kyx - u * kyz) * iw, dv1 = (kyy - v * kyz) * iw;
        const float du2 = (t2x - u * t2z) * iw, dv2 = (t2y - v * t2z) * iw;

        const float uf = floorf(u), vf = floorf(v);
        const float fu = u - uf, fv = v - vf;
        const int u0 = (int)uf, v0 = (int)vf;       // in range by validity
        const int u1 = min(u0 + 1, WG - 1);
        const int v1 = min(v0 + 1, HG - 1);
        const int o00 = v0 * WG + u0, o01 = v0 * WG + u1;
        const int o10 = v1 * WG + u0, o11 = v1 * WG + u1;
        const float w00 = (1.f - fu) * (1.f - fv), w01 = fu * (1.f - fv);
        const float w10 = (1.f - fu) * fv,         w11 = fu * fv;

        const float* cb = conf + b * (HG * WG);
        const float  cf = cb[o00] * w00 + cb[o01] * w01 + cb[o10] * w10 + cb[o11] * w11;

        const float* gb = grd + (size_t)b * CCH * HG * WG;
        const float* sb = sat + (((size_t)b * CCH) * A_SIDE + i) * A_SIDE + j;

        float Suu = 0.f, Suv = 0.f, Svv = 0.f, Sur = 0.f, Svr = 0.f;
#pragma unroll 4
        for (int c = 0; c < CCH; ++c) {
            const float* g = gb + c * (HG * WG);
            const float f00 = g[o00], f01 = g[o01], f10 = g[o10], f11 = g[o11];
            const float f    = f00 * w00 + f01 * w01 + f10 * w10 + f11 * w11;
            const float dfdu = (f01 - f00) * (1.f - fv) + (f11 - f10) * fv;
            const float dfdv = (f10 - f00) * (1.f - fu) + (f11 - f01) * fu;
            const float r    = f - sb[c * (A_SIDE * A_SIDE)];
            Suu = fmaf(dfdu, dfdu, Suu);
            Suv = fmaf(dfdu, dfdv, Suv);
            Svv = fmaf(dfdv, dfdv, Svv);
            Sur = fmaf(dfdu, r, Sur);
            Svr = fmaf(dfdv, r, Svr);
        }
        // Hs_pixel = cf * D * S2 * D^T ; rhs_pixel = cf * D * (Sur,Svr)
        const float t00 = du0 * Suu + dv0 * Suv, t01 = du0 * Suv + dv0 * Svv;
        const float t10 = du1 * Suu + dv1 * Suv, t11 = du1 * Suv + dv1 * Svv;
        const float t20 = du2 * Suu + dv2 * Suv, t21 = du2 * Suv + dv2 * Svv;
        acc[0] = cf * (t00 * du0 + t01 * dv0);
        acc[1] = cf * (t00 * du1 + t01 * dv1);
        acc[2] = cf * (t00 * du2 + t01 * dv2);
        acc[3] = cf * (t10 * du1 + t11 * dv1);
        acc[4] = cf * (t10 * du2 + t11 * dv2);
        acc[5] = cf * (t20 * du2 + t21 * dv2);
        acc[6] = cf * (du0 * Sur + dv0 * Svr);
        acc[7] = cf * (du1 * Sur + dv1 * Svr);
        acc[8] = cf * (du2 * Sur + dv2 * Svr);
    }

    // ---- block reduction: shfl within wave, WMMA across the 8 waves ----
    __shared__ float wpart[8][9];
#pragma unroll
    for (int q = 0; q < 9; ++q) {
        const float s = wave_sum32(acc[q]);
        if ((tid & 31) == 0) wpart[tid >> 5][q] = s;
    }
    __syncthreads();

    if (tid < 32) {               // wave 0, uniform condition -> EXEC all ones
        const int lane = tid;
        const int q = lane & 15;
        const bool lo = lane < 16;
        float a0 = 0.f, a1 = 0.f;
        if (q < 9) {
            a0 = lo ? (wpart[0][q] + wpart[4][q]) : (wpart[2][q] + wpart[6][q]);
            a1 = lo ? (wpart[1][q] + wpart[5][q]) : (wpart[3][q] + wpart[7][q]);
        }
        // A(16x4): row m=q holds the 4 folded partials; B = ones(4x16)
        // => D[m,n] = row-sum of A = block total for accumulator q=m.
        v2f Am; Am[0] = a0; Am[1] = a1;
        v2f Bm; Bm[0] = 1.f; Bm[1] = 1.f;
        v8f Cm = {0.f, 0.f, 0.f, 0.f, 0.f, 0.f, 0.f, 0.f};
        v8f D = __builtin_amdgcn_wmma_f32_16x16x4_f32(
            /*neg_a=*/false, Am, /*neg_b=*/false, Bm,
            /*c_mod=*/(short)0, Cm, /*reuse_a=*/false, /*reuse_b=*/false);
        float* outp = partial + ((size_t)b * NBLK + blk) * 9;
        if (lane == 0) {   // rows 0..7 live in D[0..7] on lanes 0-15
            outp[0] = D[0]; outp[1] = D[1]; outp[2] = D[2]; outp[3] = D[3];
            outp[4] = D[4]; outp[5] = D[5]; outp[6] = D[6]; outp[7] = D[7];
        }
        if (lane == 16)    // row 8 lives in D[0] on lanes 16-31
            outp[8] = D[0];
    }
}

__global__ __launch_bounds__(32)
void lm_solve(const float* __restrict__ partial, const float* __restrict__ damping,
              float* __restrict__ params, float* __restrict__ out, int write_out) {
    const int b = blockIdx.x;
    const int lane = threadIdx.x;
    __shared__ float hs[9];
    if (lane < 9) {
        float s = 0.f;
        const float* pp = partial + (size_t)b * NBLK * 9 + lane;
        for (int k = 0; k < NBLK; ++k) s += pp[k * 9];   // fixed order: deterministic
        hs[lane] = s;
    }
    __syncthreads();
    if (lane == 0) {
        const float M00 = hs[0] + damping[0];
        const float M01 = hs[1], M02 = hs[2];
        const float M11 = hs[3] + damping[1];
        const float M12 = hs[4];
        const float M22 = hs[5] + damping[2];
        const float b0 = hs[6], b1 = hs[7], b2 = hs[8];
        const float c00 = M11 * M22 - M12 * M12;
        const float c01 = M01 * M22 - M12 * M02;
        const float c02 = M01 * M12 - M11 * M02;
        const float det = M00 * c00 - M01 * c01 + M02 * c02;
        const float inv = 1.0f / det;
        const float dx = (b0 * c00 - M01 * (b1 * M22 - M12 * b2) + M02 * (b1 * M12 - M11 * b2)) * inv;
        const float dy = (M00 * (b1 * M22 - M12 * b2) - b0 * c01 + M02 * (M01 * b2 - b1 * M02)) * inv;
        const float dz = (M00 * (M11 * b2 - b1 * M12) - M01 * (M01 * b2 - b1 * M02) + b0 * c02) * inv;
        const float p0 = params[b * 3 + 0] - dx;   // delta = -solve(...)
        const float p1 = params[b * 3 + 1] - dy;
        const float p2 = params[b * 3 + 2] - dz;
        params[b * 3 + 0] = p0;
        params[b * 3 + 1] = p1;
        params[b * 3 + 2] = p2;
        if (write_out) {
            out[b * 3 + 0] = p0;
            out[b * 3 + 1] = p1;
            out[b * 3 + 2] = p2;
        }
    }
}

extern "C" void kernel_launch(void* const* d_in, const int* in_sizes, int n_in,
                              void* d_out, int out_size, void* d_ws, size_t ws_size,
                              hipStream_t stream) {
    (void)in_sizes; (void)n_in; (void)out_size; (void)ws_size;
    const float* sat     = (const float*)d_in[0];
    const float* grd     = (const float*)d_in[1];
    const float* conf    = (const float*)d_in[2];
    const float* camk    = (const float*)d_in[3];
    const float* su      = (const float*)d_in[4];
    const float* sv      = (const float*)d_in[5];
    const float* hd      = (const float*)d_in[6];
    const float* damping = (const float*)d_in[7];
    float* out    = (float*)d_out;
    float* params = (float*)d_ws;          // 24 floats
    float* partial = params + 32;          // 8*64*9 floats

    lm_init<<<1, 32, 0, stream>>>(su, sv, hd, params);
    dim3 grid(NBLK, BATCH);
    for (int it = 0; it < 2; ++it) {
        lm_accum<<<grid, PBLK, 0, stream>>>(sat, grd, conf, camk, params, partial);
        lm_solve<<<BATCH, 32, 0, stream>>>(partial, damping, params, out, it == 1);
    }
}